// ExtremeMemoryEfficientBlock_12068858102139
// MI455X (gfx1250) — compile-verified
//
#include <hip/hip_runtime.h>
#include <math.h>

// ---------------------------------------------------------------------------
// MI455X (gfx1250) fused low-rank block:
//   RMSNorm -> u=xn@U -> linear scan -> y=h@V -> +res -> RMSNorm -> FFN(lowrank,gelu) -> +res
// All matmuls use V_WMMA_F32_16X16X4_F32 (full fp32, 2048 flop/instr): the
// workload is ~3.6 GFLOP vs 268MB minimal HBM traffic; on VALU fp32 it would
// be compute-bound (~4x the 11.5us HBM time), WMMA pulls it under the roof.
// Round-2 changes: branchless (clamp+mask) padded B loads to kill per-load
// exec-mask churn in the hot WMMA loops; norm2_w staged in LDS; float4 loads
// in kernel 1.
// ---------------------------------------------------------------------------

typedef float v2f __attribute__((ext_vector_type(2)));
typedef float v8f __attribute__((ext_vector_type(8)));

#define B_     4
#define S_     4096
#define D_     2048
#define R_     6
#define RP_    8        // rank padded to 8 for WMMA K (2 k-steps of 4)
#define RF_    21
#define RFP_   24       // ffn rank padded to 24 (6 k-steps of 4)
#define NTOK   (B_*S_)
#define CHUNK  512
#define NCHUNK (S_/CHUNK)
#define EPSF   1e-6f
#define XS_STRIDE 2052  // 2048+4: half-wave row groups land on disjoint LDS banks

__device__ __forceinline__ float sigmoidf_(float x){ return 1.f/(1.f+__expf(-x)); }
__device__ __forceinline__ float geluf_(float x){
  float x3 = x*x*x;
  return 0.5f*x*(1.f + tanhf(0.7978845608028654f*(x + 0.044715f*x3)));
}

// ---------------------------------------------------------------------------
// Kernel 1: RMSNorm1 + input projection u = (w1*x/rms) @ U   (one wave per token)
// Single pass: sum-of-squares and the 6 rank dots accumulate together; the
// 1/rms scale is applied after the cross-lane reduction. float4 x/w loads.
// ---------------------------------------------------------------------------
__global__ __launch_bounds__(256) void k_norm1_uproj(
    const float* __restrict__ x, const float* __restrict__ w1n,
    const float* __restrict__ U, float* __restrict__ ubuf)
{
  int wv = threadIdx.x >> 5, lane = threadIdx.x & 31;
  int tk = blockIdx.x*8 + wv;
  const float4* xr = (const float4*)(x + (size_t)tk * D_);
  const float4* wr = (const float4*)w1n;
  float ss = 0.f;
  float acc[R_] = {0.f,0.f,0.f,0.f,0.f,0.f};
  for (int i=0;i<D_/128;i++){
    int q = lane + 32*i;                 // float4 index
    float4 xv = xr[q];
    float4 wv4 = wr[q];
    ss += xv.x*xv.x + xv.y*xv.y + xv.z*xv.z + xv.w*xv.w;
    float xw[4] = { xv.x*wv4.x, xv.y*wv4.y, xv.z*wv4.z, xv.w*wv4.w };
    int d0 = q*4;
    #pragma unroll
    for (int e=0;e<4;e++){
      const v2f* Ur = (const v2f*)(U + (d0+e)*R_);  // 24B row, 8B-aligned
      v2f u0 = Ur[0], u1 = Ur[1], u2 = Ur[2];
      acc[0] += xw[e]*u0.x; acc[1] += xw[e]*u0.y;
      acc[2] += xw[e]*u1.x; acc[3] += xw[e]*u1.y;
      acc[4] += xw[e]*u2.x; acc[5] += xw[e]*u2.y;
    }
  }
  #pragma unroll
  for (int off=16; off>0; off>>=1){
    ss += __shfl_xor(ss, off, 32);
    #pragma unroll
    for (int r=0;r<R_;r++) acc[r] += __shfl_xor(acc[r], off, 32);
  }
  if (lane==0){
    float rr = rsqrtf(ss*(1.f/(float)D_) + EPSF);
    #pragma unroll
    for (int r=0;r<R_;r++) ubuf[tk*RP_ + r] = acc[r]*rr;
  }
}

// ---------------------------------------------------------------------------
// Kernel 2: per-chunk Kogge-Stone scan of h_t = a*h_{t-1} + u_t.
// Associative op on (A,u): (A2,u2) o (A1,u1) = (A1*A2, A2*u1 + u2).
// ---------------------------------------------------------------------------
__global__ __launch_bounds__(CHUNK) void k_scan_local(
    const float* __restrict__ ubuf, const float* __restrict__ lam,
    float* __restrict__ hbuf, float* __restrict__ cend, float* __restrict__ cendA)
{
  __shared__ float Ut[R_][CHUNK];
  __shared__ float At[R_][CHUNK];
  int t = threadIdx.x;
  int b = blockIdx.x / NCHUNK, c = blockIdx.x % NCHUNK;
  int tok = b*S_ + c*CHUNK + t;
  #pragma unroll
  for (int r=0;r<R_;r++){
    Ut[r][t] = ubuf[tok*RP_ + r];
    At[r][t] = sigmoidf_(lam[r]);
  }
  __syncthreads();
  for (int off=1; off<CHUNK; off<<=1){
    float pu[R_], pa[R_], cu[R_], ca[R_];
    #pragma unroll
    for (int r=0;r<R_;r++){
      cu[r]=Ut[r][t]; ca[r]=At[r][t];
      pu[r] = (t>=off)? Ut[r][t-off] : 0.f;
      pa[r] = (t>=off)? At[r][t-off] : 1.f;
    }
    __syncthreads();
    #pragma unroll
    for (int r=0;r<R_;r++){
      Ut[r][t] = ca[r]*pu[r] + cu[r];
      At[r][t] = ca[r]*pa[r];
    }
    __syncthreads();
  }
  #pragma unroll
  for (int r=0;r<R_;r++) hbuf[tok*RP_ + r] = Ut[r][t];
  hbuf[tok*RP_ + 6] = 0.f;     // WMMA K-padding
  hbuf[tok*RP_ + 7] = 0.f;
  if (t==CHUNK-1){
    int ci = b*NCHUNK + c;
    #pragma unroll
    for (int r=0;r<R_;r++){ cend[ci*RP_+r] = Ut[r][t]; cendA[ci*RP_+r] = At[r][t]; }
  }
}

// ---------------------------------------------------------------------------
// Kernel 3: sequential scan of chunk carries (24 chains x 8 chunks; trivial).
// ---------------------------------------------------------------------------
__global__ void k_scan_carry(const float* __restrict__ cend,
                             const float* __restrict__ cendA,
                             float* __restrict__ carry)
{
  int tid = threadIdx.x;
  if (tid >= B_*R_) return;
  int b = tid / R_, r = tid % R_;
  float cur = 0.f;
  for (int c=0;c<NCHUNK;c++){
    int ci = b*NCHUNK + c;
    carry[ci*RP_ + r] = cur;                       // carry entering chunk c
    cur = cendA[ci*RP_+r]*cur + cend[ci*RP_+r];
  }
}

// ---------------------------------------------------------------------------
// Kernel 4: h[t] = local[t] + a^(tl+1) * carry_in(chunk)
// ---------------------------------------------------------------------------
__global__ __launch_bounds__(256) void k_apply_carry(
    float* __restrict__ hbuf, const float* __restrict__ carry,
    const float* __restrict__ lam)
{
  int t = blockIdx.x*256 + threadIdx.x;
  int b = t / S_, s = t % S_;
  int c = s / CHUNK, tl = s % CHUNK;
  int ci = b*NCHUNK + c;
  float e = (float)(tl+1);
  #pragma unroll
  for (int r=0;r<R_;r++){
    float a = sigmoidf_(lam[r]);                   // in (0,1): log is safe
    float p = __expf(e*__logf(a));
    hbuf[t*RP_+r] += p*carry[ci*RP_+r];
  }
}

// ---------------------------------------------------------------------------
// Kernel 5: all post-scan work is row-local -> fully fused per 16-token tile.
// 256 threads = 8 waves; x tile lives in LDS (x is L2-resident on reread).
// WMMA fragment conventions (f32 16x16x4): A[16x4]: lane<16 -> M, VGPR pair =
// K{2*half,2*half+1}; B[4x16]: lane%16 -> N (halves mirrored over K); C/D:
// VGPR j holds M=j (+8 for high half-wave), lane%16 = N.
// All padded loads are branchless: clamped address * loop-invariant mask.
// ---------------------------------------------------------------------------
__global__ __launch_bounds__(256) void k_fused_out(
    const float* __restrict__ x, const float* __restrict__ hbuf,
    const float* __restrict__ V, const float* __restrict__ w2n,
    const float* __restrict__ W1, const float* __restrict__ W2,
    float* __restrict__ out)
{
  __shared__ __align__(16) float xs[16*XS_STRIDE];  // x -> x2 -> out tile
  __shared__ __align__(16) float wns[D_];           // norm2_w staged in LDS
  __shared__ float hsT[16*RP_];
  __shared__ float rrms[16];
  __shared__ float gs[16*RFP_];
  __shared__ float part[8*2*32*8];                  // per-wave partial C tiles

  int wv = threadIdx.x>>5, lane = threadIdx.x&31;
  int tok0 = blockIdx.x*16;
  int mlo = lane & 15, khf = lane >> 4;

  // --- stage a: cooperative x tile + norm2_w load (float4) ------------------
  for (int rr=0; rr<2; rr++){
    int row = wv*2 + rr;
    const float4* src = (const float4*)(x + (size_t)(tok0+row)*D_);
    float4* dst = (float4*)(xs + row*XS_STRIDE);
    for (int i=lane;i<D_/4;i+=32) dst[i] = src[i];
  }
  {
    const float4* src = (const float4*)w2n;
    float4* dst = (float4*)wns;
    for (int i=threadIdx.x;i<D_/4;i+=256) dst[i] = src[i];
  }
  if (threadIdx.x < 16*RP_) hsT[threadIdx.x] = hbuf[(size_t)tok0*RP_ + threadIdx.x];
  __syncthreads();

  // --- stage b: y = h @ V (K=8 -> 2 WMMA per 16x16 tile), x2 = x + y --------
  {
    v2f a0, a1;
    a0.x = hsT[mlo*RP_ + 2*khf];       a0.y = hsT[mlo*RP_ + 2*khf+1];
    a1.x = hsT[mlo*RP_ + 4 + 2*khf];   a1.y = hsT[mlo*RP_ + 4 + 2*khf+1];
    int k  = 2*khf;                                   // rows 0/1 or 2/3: valid
    int k2 = 4 + 2*khf;                               // rows 4/5 or 6/7
    float m2x = (k2   < R_) ? 1.f : 0.f;
    float m2y = (k2+1 < R_) ? 1.f : 0.f;
    int k2cx = (k2   < R_) ? k2   : 0;
    int k2cy = (k2+1 < R_) ? k2+1 : 0;
    for (int ti=0; ti<16; ti++){
      int n = (wv + 8*ti)*16 + mlo;
      v2f b0, b1; v8f c = {};
      b0.x = V[k*D_ + n];       b0.y = V[(k+1)*D_ + n];
      b1.x = V[k2cx*D_ + n]*m2x; b1.y = V[k2cy*D_ + n]*m2y;
      c = __builtin_amdgcn_wmma_f32_16x16x4_f32(false, a0, false, b0, (short)0, c, false, false);
      c = __builtin_amdgcn_wmma_f32_16x16x4_f32(false, a1, false, b1, (short)0, c, false, false);
      #pragma unroll
      for (int j=0;j<8;j++){
        int m = j + 8*khf;
        xs[m*XS_STRIDE + n] += c[j];
      }
    }
  }
  __syncthreads();

  // --- stage c: RMSNorm2 row stats ------------------------------------------
  for (int rr=0; rr<2; rr++){
    int row = wv*2+rr;
    float ss=0.f;
    for (int d=lane; d<D_; d+=32){ float v = xs[row*XS_STRIDE+d]; ss += v*v; }
    #pragma unroll
    for (int off=16;off>0;off>>=1) ss += __shfl_xor(ss, off, 32);
    if (lane==0) rrms[row] = rsqrtf(ss*(1.f/(float)D_)+EPSF);
  }
  __syncthreads();

  // --- stage d: t = xn2 @ W1, K split over waves, reduce in LDS, gelu -------
  {
    int kbase = wv*256;
    v8f c0 = {}, c1 = {};
    float rm = rrms[mlo];
    int j1  = 16 + mlo;
    int j1c = (j1<RF_) ? j1 : 0;                      // branchless pad
    float m1 = (j1<RF_) ? 1.f : 0.f;
    for (int ks=0; ks<64; ks++){
      int k = kbase + ks*4 + 2*khf;
      v2f a, b0, b1;
      a.x = xs[mlo*XS_STRIDE + k]   * wns[k]   * rm;
      a.y = xs[mlo*XS_STRIDE + k+1] * wns[k+1] * rm;
      b0.x = W1[k*RF_ + mlo];       b0.y = W1[(k+1)*RF_ + mlo];
      b1.x = W1[k*RF_ + j1c]*m1;    b1.y = W1[(k+1)*RF_ + j1c]*m1;
      c0 = __builtin_amdgcn_wmma_f32_16x16x4_f32(false, a, false, b0, (short)0, c0, false, false);
      c1 = __builtin_amdgcn_wmma_f32_16x16x4_f32(false, a, false, b1, (short)0, c1, false, false);
    }
    float* pb = part + ((wv*2+0)*32 + lane)*8;
    #pragma unroll
    for (int j=0;j<8;j++) pb[j] = c0[j];
    pb = part + ((wv*2+1)*32 + lane)*8;
    #pragma unroll
    for (int j=0;j<8;j++) pb[j] = c1[j];
  }
  __syncthreads();
  for (int idx = threadIdx.x; idx < 16*RFP_; idx += 256){
    int m = idx / RFP_, j = idx % RFP_;
    int tile = (j>=16) ? 1 : 0, n = j & 15;
    int lsel = n + 16*(m>>3), vg = m & 7;
    float s = 0.f;
    #pragma unroll
    for (int w=0;w<8;w++) s += part[((w*2+tile)*32+lsel)*8+vg];
    gs[m*RFP_+j] = (j<RF_) ? geluf_(s) : 0.f;
  }
  __syncthreads();

  // --- stage e: out_tile += gelu(t) @ W2 (K=24 -> 6 WMMA per tile) ----------
  {
    v2f af[6];
    int  jc [6]; float mx[6];
    int  jc1[6]; float my[6];
    #pragma unroll
    for (int k0=0;k0<6;k0++){
      int j = 4*k0 + 2*khf;
      af[k0].x = gs[mlo*RFP_ + j];
      af[k0].y = gs[mlo*RFP_ + j + 1];
      jc [k0] = (j   < RF_) ? j   : 0;  mx[k0] = (j   < RF_) ? 1.f : 0.f;
      jc1[k0] = (j+1 < RF_) ? j+1 : 0;  my[k0] = (j+1 < RF_) ? 1.f : 0.f;
    }
    for (int ti=0; ti<16; ti++){
      int n = (wv + 8*ti)*16 + mlo;
      v8f c = {};
      #pragma unroll
      for (int k0=0;k0<6;k0++){
        v2f b;
        b.x = W2[jc [k0]*D_ + n]*mx[k0];
        b.y = W2[jc1[k0]*D_ + n]*my[k0];
        c = __builtin_amdgcn_wmma_f32_16x16x4_f32(false, af[k0], false, b, (short)0, c, false, false);
      }
      #pragma unroll
      for (int jj=0;jj<8;jj++){
        int m = jj + 8*khf;
        xs[m*XS_STRIDE + n] += c[jj];
      }
    }
  }
  __syncthreads();

  // --- stage f: coalesced store ---------------------------------------------
  for (int rr=0;rr<2;rr++){
    int row = wv*2+rr;
    const float4* srcl = (const float4*)(xs + row*XS_STRIDE);
    float4* dst = (float4*)(out + (size_t)(tok0+row)*D_);
    for (int i=lane;i<D_/4;i+=32) dst[i] = srcl[i];
  }
}

// ---------------------------------------------------------------------------
extern "C" void kernel_launch(void* const* d_in, const int* in_sizes, int n_in,
                              void* d_out, int out_size, void* d_ws, size_t ws_size,
                              hipStream_t stream) {
  (void)in_sizes; (void)n_in; (void)out_size; (void)ws_size;
  const float* x   = (const float*)d_in[0];
  const float* n1w = (const float*)d_in[1];
  const float* U   = (const float*)d_in[2];
  const float* lam = (const float*)d_in[3];
  const float* V   = (const float*)d_in[4];
  const float* n2w = (const float*)d_in[5];
  const float* W1  = (const float*)d_in[6];
  const float* W2  = (const float*)d_in[7];
  float* out = (float*)d_out;

  char* ws = (char*)d_ws;
  float* ubuf  = (float*)(ws);                      // [NTOK][8]  512 KB
  float* hbuf  = (float*)(ws + 524288);             // [NTOK][8]  512 KB
  float* cend  = (float*)(ws + 1048576);            // [B*NCHUNK][8] 1 KB
  float* cendA = (float*)(ws + 1048576 + 4096);     // [B*NCHUNK][8] 1 KB
  float* carry = (float*)(ws + 1048576 + 8192);     // [B*NCHUNK][8] 1 KB

  k_norm1_uproj<<<NTOK/8, 256, 0, stream>>>(x, n1w, U, ubuf);
  k_scan_local<<<B_*NCHUNK, CHUNK, 0, stream>>>(ubuf, lam, hbuf, cend, cendA);
  k_scan_carry<<<1, 32, 0, stream>>>(cend, cendA, carry);
  k_apply_carry<<<NTOK/256, 256, 0, stream>>>(hbuf, carry, lam);
  k_fused_out<<<NTOK/16, 256, 0, stream>>>(x, hbuf, V, n2w, W1, W2, out);
}